// GraphSAGE_33432025432295
// MI455X (gfx1250) — compile-verified
//
#include <hip/hip_runtime.h>

typedef __attribute__((ext_vector_type(2))) float v2f;
typedef __attribute__((ext_vector_type(8))) float v8f;

// ---------------------------------------------------------------------------
// Zero-fill scratch (atomic accumulators must start at 0 every call).
// ---------------------------------------------------------------------------
__global__ void zero_f32(float* __restrict__ p, size_t n) {
    size_t i = (size_t)blockIdx.x * blockDim.x + threadIdx.x;
    size_t i4 = i * 4;
    if (i4 + 3 < n) {
        *(float4*)(p + i4) = make_float4(0.f, 0.f, 0.f, 0.f);
    } else {
        for (size_t k = i4; k < n; ++k) p[k] = 0.0f;
    }
}

// ---------------------------------------------------------------------------
// Repack a 128x128 row-major weight matrix into WMMA B-fragment order so the
// GEMM does one coalesced global_load_b64 per fragment:
//   P[(((t*8 + n)*32 + L)*2 + {0,1}] = W[4t + 2*(L>=16) + {0,1}][n*16 + (L&15)]
// ---------------------------------------------------------------------------
__global__ __launch_bounds__(256) void repack_w(const float* __restrict__ W,
                                                float* __restrict__ P) {
    const int i = blockIdx.x * blockDim.x + threadIdx.x;   // 0 .. 8191
    if (i >= 32 * 8 * 32) return;
    const int L = i & 31;
    const int n = (i >> 5) & 7;
    const int t = i >> 8;
    const int half = L >> 4;
    const int l16 = L & 15;
    const int ka = 4 * t + 2 * half;
    const int col = n * 16 + l16;
    float2 v;
    v.x = W[(size_t)ka * 128 + col];
    v.y = W[(size_t)(ka + 1) * 128 + col];
    *(float2*)(P + (size_t)i * 2) = v;
}

// ---------------------------------------------------------------------------
// Edge-parallel scatter: agg[dst] += feat[src]; cnt[dst] += 1 (layer 1 only).
// One wave32 per edge, 4 floats per lane (D = 128). Atomics land in the
// L2-resident accumulator (51 MB << 192 MB L2).
// ---------------------------------------------------------------------------
__global__ __launch_bounds__(256) void sage_scatter(
    const float* __restrict__ feat,      // [N,128]
    const long long* __restrict__ src,   // [E]
    const long long* __restrict__ dst,   // [E]
    float* __restrict__ agg,             // [N,128]
    float* __restrict__ cnt,             // [N] or nullptr
    int E)
{
    const int lane = threadIdx.x & 31;
    const int e = blockIdx.x * 8 + (threadIdx.x >> 5);
    if (e >= E) return;
    const long long s = src[e];
    const long long d = dst[e];
    const float4 v = *(const float4*)(feat + s * 128 + lane * 4);
    float* ap = agg + d * 128 + lane * 4;
    atomicAdd(ap + 0, v.x);
    atomicAdd(ap + 1, v.y);
    atomicAdd(ap + 2, v.z);
    atomicAdd(ap + 3, v.w);
    if (cnt != nullptr && lane == 0) atomicAdd(cnt + d, 1.0f);
}

// ---------------------------------------------------------------------------
// Fused SAGE layer: out = act( (agg/max(cnt,1)) @ Wl + xin @ Wr + bias )
// V_WMMA_F32_16X16X4_F32. Block = 256 threads = 8 waves; each wave owns a
// 16-row x 128-col output tile (8 accumulator tiles of 16x16).
// WlP / WrP are fragment-order repacked weights (see repack_w).
// ---------------------------------------------------------------------------
template <bool RELU>
__global__ __launch_bounds__(256) void sage_gemm(
    const float* __restrict__ agg,   // [N,128] summed neighbor features
    const float* __restrict__ cnt,   // [N]
    const float* __restrict__ xin,   // [N,128] self features
    const float* __restrict__ WlP,   // [32*8*32*2] packed
    const float* __restrict__ WrP,   // [32*8*32*2] packed
    const float* __restrict__ bias,  // [128]
    float* __restrict__ out,         // [N,128]
    int N)
{
    const int lane = threadIdx.x & 31;
    const int wave = threadIdx.x >> 5;
    const int half = lane >> 4;        // 0: lanes 0-15, 1: lanes 16-31
    const int l16  = lane & 15;
    const int row0 = blockIdx.x * 128 + wave * 16;

    // Row this lane feeds into A fragments. Clamp (don't branch) so EXEC
    // stays all-1s through the WMMA region; stores handle bounds later.
    int arow = row0 + l16;
    if (arow >= N) arow = N - 1;

    const float inv = 1.0f / fmaxf(cnt[arow], 1.0f);
    const float* aggRow = agg + (size_t)arow * 128;
    const float* xRow   = xin + (size_t)arow * 128;

    // Per-lane base into the packed weight stream; step t advances by 8*32*2.
    const float* blp = WlP + (size_t)lane * 2;
    const float* brp = WrP + (size_t)lane * 2;

    v8f acc[8];
#pragma unroll
    for (int n = 0; n < 8; ++n) acc[n] = (v8f)0.0f;

#pragma unroll 2
    for (int t = 0; t < 32; ++t) {
        const int ka = 4 * t + half * 2;   // this lane's K base for A frags

        __builtin_prefetch(aggRow + ka + 16, 0, 0);
        __builtin_prefetch(xRow + ka + 16, 0, 0);

        v2f a_mean, a_self;
        {
            const float2 tv = *(const float2*)(aggRow + ka);
            a_mean[0] = tv.x * inv;
            a_mean[1] = tv.y * inv;
            const float2 u = *(const float2*)(xRow + ka);
            a_self[0] = u.x;
            a_self[1] = u.y;
        }

        const float* blt = blp + (size_t)t * (8 * 32 * 2);
        const float* brt = brp + (size_t)t * (8 * 32 * 2);
#pragma unroll
        for (int n = 0; n < 8; ++n) {
            const v2f bl = *(const v2f*)(blt + n * 64);   // coalesced b64
            const v2f br = *(const v2f*)(brt + n * 64);
            acc[n] = __builtin_amdgcn_wmma_f32_16x16x4_f32(
                false, a_mean, false, bl, (short)0, acc[n], false, false);
            acc[n] = __builtin_amdgcn_wmma_f32_16x16x4_f32(
                false, a_self, false, br, (short)0, acc[n], false, false);
        }
    }

    // Epilogue: bias (+ReLU). Wave-uniform bounds check -> fast path has
    // zero per-lane predication.
    if (row0 + 16 <= N) {
#pragma unroll
        for (int n = 0; n < 8; ++n) {
            const int col = n * 16 + l16;
            const float b = bias[col];
#pragma unroll
            for (int j = 0; j < 8; ++j) {
                float v = acc[n][j] + b;
                if (RELU) v = fmaxf(v, 0.0f);
                out[(size_t)(row0 + j + 8 * half) * 128 + col] = v;
            }
        }
    } else {
#pragma unroll
        for (int n = 0; n < 8; ++n) {
            const int col = n * 16 + l16;
            const float b = bias[col];
#pragma unroll
            for (int j = 0; j < 8; ++j) {
                const int r = row0 + j + 8 * half;
                if (r < N) {
                    float v = acc[n][j] + b;
                    if (RELU) v = fmaxf(v, 0.0f);
                    out[(size_t)r * 128 + col] = v;
                }
            }
        }
    }
}

// ---------------------------------------------------------------------------
// Host-side launch. Workspace layout (floats):
//   agg1 [N*128] | agg2 [N*128] | cnt [N] | h [N*128] | 4x packed W [16384]
// agg1|agg2|cnt zeroed every call (contiguous region).
// ---------------------------------------------------------------------------
extern "C" void kernel_launch(void* const* d_in, const int* in_sizes, int n_in,
                              void* d_out, int out_size, void* d_ws, size_t ws_size,
                              hipStream_t stream) {
    const float*      x   = (const float*)d_in[0];
    const long long*  ei  = (const long long*)d_in[1];
    const float*      Wl1 = (const float*)d_in[2];
    const float*      Wr1 = (const float*)d_in[3];
    const float*      b1  = (const float*)d_in[4];
    const float*      Wl2 = (const float*)d_in[5];
    const float*      Wr2 = (const float*)d_in[6];
    const float*      b2  = (const float*)d_in[7];
    float*            out = (float*)d_out;

    const int N = in_sizes[0] / 128;
    const int E = in_sizes[1] / 2;
    const size_t ND = (size_t)N * 128;
    const size_t WP = 32 * 8 * 32 * 2;   // 16384 floats per packed matrix

    float* agg1 = (float*)d_ws;
    float* agg2 = agg1 + ND;
    float* cnt  = agg2 + ND;
    float* h    = cnt  + (size_t)N;
    float* Wl1P = h    + ND;
    float* Wr1P = Wl1P + WP;
    float* Wl2P = Wr1P + WP;
    float* Wr2P = Wl2P + WP;

    const long long* src = ei;        // edge_index[0]
    const long long* dst = ei + E;    // edge_index[1]

    // 1) zero accumulators (agg1|agg2|cnt contiguous)
    {
        size_t zn = ND * 2 + (size_t)N;
        int blocks = (int)((zn / 4 + 255) / 256) + 1;
        zero_f32<<<blocks, 256, 0, stream>>>(agg1, zn);
    }

    // 2) repack weights into WMMA B-fragment order (one-shot, 64 KB each)
    repack_w<<<32, 256, 0, stream>>>(Wl1, Wl1P);
    repack_w<<<32, 256, 0, stream>>>(Wr1, Wr1P);
    repack_w<<<32, 256, 0, stream>>>(Wl2, Wl2P);
    repack_w<<<32, 256, 0, stream>>>(Wr2, Wr2P);

    const int scatter_blocks = (E + 7) / 8;
    const int gemm_blocks    = (N + 127) / 128;

    // 3) layer-1 scatter (also builds cnt, reused by layer 2)
    sage_scatter<<<scatter_blocks, 256, 0, stream>>>(x, src, dst, agg1, cnt, E);

    // 4) layer-1 fused mean-GEMM + bias + ReLU -> h
    sage_gemm<true><<<gemm_blocks, 256, 0, stream>>>(agg1, cnt, x, Wl1P, Wr1P, b1, h, N);

    // 5) layer-2 scatter from h
    sage_scatter<<<scatter_blocks, 256, 0, stream>>>(h, src, dst, agg2, nullptr, E);

    // 6) layer-2 fused mean-GEMM + bias -> out
    sage_gemm<false><<<gemm_blocks, 256, 0, stream>>>(agg2, cnt, h, Wl2P, Wr2P, b2, out, N);
}